// Compact_Triplet_MMD_scr_39926015983686
// MI455X (gfx1250) — compile-verified
//
#include <hip/hip_runtime.h>
#include <hip/hip_bf16.h>
#include <math.h>

// ---------------- problem constants (from reference) ----------------
#define T_LEN   512
#define DFEAT   64
#define MFEAT   (T_LEN * DFEAT)     // 32768 flattened features per sequence
#define NWIN    4
#define NGAL    5
#define NNEG    10
#define STEPW   16                  // NG + NF + 1
#define NPAIR   60                  // NW * (NG + NF)
#define GAMMA_F 5.0f
#define BIGV    1e10f
#define MMD_N   11                  // STEP - half
#define MMD_M   22                  // 2 * MMD_N
#define MMD_PAIRS 9

typedef __attribute__((ext_vector_type(2))) float v2f;
typedef __attribute__((ext_vector_type(8))) float v8f;

// MMD window pairs: i in 0..3, j in 1..3, j != i (reference order)
__constant__ int c_mmd_i[MMD_PAIRS] = {0,0,0,1,1,2,2,3,3};
__constant__ int c_mmd_j[MMD_PAIRS] = {1,2,3,2,3,1,3,1,2};

// ---------------- kernel 0: per-row squared norms -------------------
__global__ void k_norms(const float* __restrict__ data, float* __restrict__ norms) {
    const int s = blockIdx.x;                       // sequence 0..63
    for (int row = threadIdx.x; row < T_LEN; row += blockDim.x) {
        const float* p = data + (size_t)s * MFEAT + row * DFEAT;
        float acc = 0.f;
        #pragma unroll
        for (int d = 0; d < DFEAT; d += 4) {
            float4 v = *(const float4*)(p + d);
            acc += v.x*v.x + v.y*v.y + v.z*v.z + v.w*v.w;
        }
        norms[s * T_LEN + row] = acc;
    }
}

// ---------------- kernel 1: cost matrices via fp32 WMMA -------------
// Dm[p][i][j] = |A_i|^2 + |B_j|^2 - 2 * <A_i, B_j>, G via V_WMMA_F32_16X16X4_F32
__global__ void __launch_bounds__(256) k_cost_wmma(const float* __restrict__ data,
                                                   const float* __restrict__ norms,
                                                   float* __restrict__ Dm) {
    const int pair = blockIdx.y;                    // 0..59
    const int w    = pair / 15;
    const int aseq = STEPW * w;
    const int bseq = STEPW * w + (pair % 15) + 1;

    const int wid  = threadIdx.x >> 5;
    const int lane = threadIdx.x & 31;
    const int tile = blockIdx.x * 8 + wid;          // 0..1023 (32x32 tiles of 16x16)
    const int m0 = (tile >> 5) << 4;
    const int n0 = (tile & 31) << 4;
    const int r  = lane & 15;
    const int h  = lane >> 4;                       // half-wave selects K pair

    const float* arow = data + (size_t)aseq * MFEAT + (m0 + r) * DFEAT + 2 * h;
    const float* brow = data + (size_t)bseq * MFEAT + (n0 + r) * DFEAT + 2 * h;

    v8f c = {};
    #pragma unroll
    for (int k = 0; k < DFEAT; k += 4) {
        v2f af = *(const v2f*)(arow + k);
        v2f bf = *(const v2f*)(brow + k);
        c = __builtin_amdgcn_wmma_f32_16x16x4_f32(false, af, false, bf,
                                                  (short)0, c, false, false);
    }

    const float* na = norms + aseq * T_LEN;
    const float  nb = norms[bseq * T_LEN + n0 + r];
    float* out = Dm + (size_t)pair * (T_LEN * T_LEN);
    #pragma unroll
    for (int v = 0; v < 8; ++v) {
        const int m = m0 + v + 8 * h;               // D-matrix VGPR layout
        out[m * T_LEN + n0 + r] = na[m] + nb - 2.0f * c[v];
    }
}

// ---------------- kernel 2: soft-DTW wavefront DP -------------------
// Critical path: 1023 barrier-separated diagonals. Costs for diagonal k+1
// are staged into LDS with GLOBAL_LOAD_ASYNC_TO_LDS_B32 (ASYNCcnt-tracked,
// completes in order) while diagonal k's softmin runs, so L2 latency is
// fully off the serial chain. Each lane reads back only its own LDS slot.
__global__ void __launch_bounds__(512) k_softdtw(const float* __restrict__ Dm,
                                                 const int* __restrict__ lens,
                                                 float* __restrict__ raw) {
    __shared__ float buf[3][T_LEN + 1];
    __shared__ float cbuf[2][T_LEN];
    const int pair = blockIdx.x;
    const int w    = pair / 15;
    const int la   = lens[STEPW * w];
    const int lb   = lens[STEPW * w + (pair % 15) + 1];
    const int ktarget = la + lb;

    const int t = threadIdx.x;
    const int i = t + 1;                            // diagonal row index 1..512
    const float* cm = Dm + (size_t)pair * (T_LEN * T_LEN);

    buf[0][i] = BIGV;                               // r0: only [0]=0
    buf[1][i] = BIGV;                               // r1: all BIG
    if (t == 0) { buf[0][0] = 0.f; buf[1][0] = BIGV; }

    // async prefetch of diagonal k=2 costs into cbuf[0]
    {
        const int j0 = 2 - i;
        const float* src = (j0 >= 1 && j0 <= T_LEN)
                         ? (cm + (size_t)(i - 1) * T_LEN + (j0 - 1)) : cm;
        const unsigned dst = (unsigned)(uintptr_t)&cbuf[0][t]; // LDS byte offset
        asm volatile("global_load_async_to_lds_b32 %0, %1, off"
                     :: "v"(dst), "v"(src) : "memory");
    }
    __syncthreads();

    const float inv_g = 1.0f / GAMMA_F;
    for (int k = 2; k <= 2 * T_LEN; ++k) {
        const float* p2 = buf[(k - 2) % 3];
        const float* p1 = buf[(k - 1) % 3];
        float* cur = buf[k % 3];

        // issue next diagonal's async cost loads, then wait for this one's
        if (k < 2 * T_LEN) {
            const int jn = k + 1 - i;
            const float* src = (jn >= 1 && jn <= T_LEN)
                             ? (cm + (size_t)(i - 1) * T_LEN + (jn - 1)) : cm;
            const unsigned dst = (unsigned)(uintptr_t)&cbuf[(k + 1) & 1][t];
            asm volatile("global_load_async_to_lds_b32 %0, %1, off"
                         :: "v"(dst), "v"(src) : "memory");
            asm volatile("s_wait_asynccnt 0x1" ::: "memory");
        } else {
            asm volatile("s_wait_asynccnt 0x0" ::: "memory");
        }

        const int j = k - i;
        float out = BIGV;
        if (j >= 1 && j <= T_LEN) {
            const float x0 = p2[i - 1], x1 = p1[i - 1], x2 = p1[i];
            const float m  = fminf(fminf(x0, x1), x2);
            const float e  = expf((m - x0) * inv_g) + expf((m - x1) * inv_g) +
                             expf((m - x2) * inv_g);
            out = cbuf[k & 1][t] + (m - GAMMA_F * logf(e));
        }
        cur[i] = out;
        if (t == 0) cur[0] = BIGV;
        if (k == ktarget && i == la) raw[pair] = out;
        __syncthreads();
    }
}

// ---------------- kernel 3: triplet-style loss on 60 scalars --------
__global__ void k_triplet(const float* __restrict__ raw, const int* __restrict__ lens,
                          float* __restrict__ loss_out) {
    if (threadIdx.x != 0 || blockIdx.x != 0) return;
    float total = 0.f;
    for (int w = 0; w < NWIN; ++w) {
        float dist[15];
        for (int c = 0; c < 15; ++c) {
            const int la = lens[STEPW * w], lb = lens[STEPW * w + c + 1];
            dist[c] = raw[w * 15 + c] / (float)(la + lb);
        }
        float mg = 0.f;
        for (int g = 0; g < NGAL; ++g) mg += dist[g];
        mg /= (float)NGAL;
        const float dk = sqrtf(fabsf(mg));
        float dg[NGAL], dn[NNEG];
        for (int g = 0; g < NGAL; ++g) dg[g] = dist[g] / dk;
        for (int n = 0; n < NNEG; ++n) dn[n] = dist[NGAL + n] / dk;
        float lk_sum = 0.f; int cnt = 0;
        for (int g = 0; g < NGAL; ++g)
            for (int n = 0; n < NNEG; ++n) {        // lk_s (n<5) + lk_r (n>=5)
                const float v = dg[g] + 1.0f /*MARGIN*/ - dn[n];
                if (v > 0.f) { lk_sum += v; ++cnt; }
            }
        float ca = 0.f, cb = 0.f, intra = 0.f;
        for (int g = 0; g < NGAL; ++g) ca += dg[g];
        ca /= (float)NGAL;
        for (int n = 0; n < 5; ++n) cb += dn[n];
        cb /= 5.0f;
        for (int g = 0; g < NGAL; ++g) intra += dg[g] - ca;
        const float inter = fmaxf(0.f, 1.0f /*BETA*/ - fabsf(ca - cb));
        total += lk_sum / ((float)cnt + 1.0f) + intra * 0.1f + inter * 0.1f;
    }
    loss_out[0] = total / (float)NWIN;
}

// ---------------- kernel 4: MMD per window-pair via fp32 WMMA -------
// gram = R*R^T with R = concat(X,Y) (22 x 32768), padded to 32x32 tiles.
// 8 waves: 4 tiles x 2 K-halves, LDS reduction, then multi-kernel MMD math.
__global__ void __launch_bounds__(256) k_mmd(const float* __restrict__ data,
                                             float* __restrict__ mmds) {
    __shared__ float part[8][256];
    __shared__ float gram[32][32];
    __shared__ float red[256];
    const int pair = blockIdx.x;
    const int si = c_mmd_i[pair], sj = c_mmd_j[pair];

    const int tid  = threadIdx.x;
    const int wid  = tid >> 5;
    const int lane = tid & 31;
    const int r = lane & 15, h = lane >> 4;
    const int tileid = wid & 3;
    const int m0 = (tileid >> 1) * 16, n0 = (tileid & 1) * 16;
    const int kbase = (wid >> 2) * (MFEAT / 2);

    const float* arow = nullptr;
    const float* brow = nullptr;
    const int mrow = m0 + r, nrow = n0 + r;
    if (mrow < MMD_M) {
        const int seq = (mrow < MMD_N) ? (STEPW * si + mrow) : (STEPW * sj + (mrow - MMD_N));
        arow = data + (size_t)seq * MFEAT + kbase + 2 * h;
    }
    if (nrow < MMD_M) {
        const int seq = (nrow < MMD_N) ? (STEPW * si + nrow) : (STEPW * sj + (nrow - MMD_N));
        brow = data + (size_t)seq * MFEAT + kbase + 2 * h;
    }

    const v2f zf = {0.f, 0.f};
    v8f c = {};
    for (int k = 0; k < MFEAT / 2; k += 4) {
        v2f af = arow ? *(const v2f*)(arow + k) : zf;
        v2f bf = brow ? *(const v2f*)(brow + k) : zf;
        c = __builtin_amdgcn_wmma_f32_16x16x4_f32(false, af, false, bf,
                                                  (short)0, c, false, false);
    }
    #pragma unroll
    for (int v = 0; v < 8; ++v)
        part[wid][(v + 8 * h) * 16 + r] = c[v];
    __syncthreads();

    for (int e = tid; e < 1024; e += 256) {         // combine the two K halves
        const int tl = e >> 8, idx = e & 255;
        const int mm = (tl >> 1) * 16 + (idx >> 4);
        const int nn = (tl & 1) * 16 + (idx & 15);
        gram[mm][nn] = part[tl][idx] + part[tl + 4][idx];
    }
    __syncthreads();

    // pass 1: sum of L2 for bandwidth
    float lsum = 0.f;
    for (int e = tid; e < MMD_M * MMD_M; e += 256) {
        const int a = e / MMD_M, b = e % MMD_M;
        lsum += gram[a][a] + gram[b][b] - 2.f * gram[a][b];
    }
    red[tid] = lsum; __syncthreads();
    for (int s = 128; s > 0; s >>= 1) { if (tid < s) red[tid] += red[tid + s]; __syncthreads(); }
    const float bw = red[0] / (float)(MMD_M * MMD_M - MMD_M) / 4.0f; // / K_MUL^(K_NUM//2)
    __syncthreads();

    // pass 2: multi-kernel sum with quadrant signs (all quadrants are 11x11)
    float ksum = 0.f;
    for (int e = tid; e < MMD_M * MMD_M; e += 256) {
        const int a = e / MMD_M, b = e % MMD_M;
        const float l2 = gram[a][a] + gram[b][b] - 2.f * gram[a][b];
        float acc = 0.f, bwq = bw;
        #pragma unroll
        for (int q = 0; q < 5; ++q) { acc += expf(-l2 / bwq); bwq *= 2.0f; }
        ksum += (((a < MMD_N) == (b < MMD_N)) ? 1.0f : -1.0f) * acc;
    }
    red[tid] = ksum; __syncthreads();
    for (int s = 128; s > 0; s >>= 1) { if (tid < s) red[tid] += red[tid + s]; __syncthreads(); }
    if (tid == 0) mmds[pair] = red[0] / (float)(MMD_N * MMD_N);
}

// ---------------- kernel 5: final combine ---------------------------
__global__ void k_final(const float* __restrict__ loss_ws, const float* __restrict__ mmds,
                        float* __restrict__ out) {
    if (threadIdx.x != 0 || blockIdx.x != 0) return;
    float mx = 0.f;                                 // zero padding participates in max
    for (int p = 0; p < MMD_PAIRS; ++p) mx = fmaxf(mx, mmds[p]);
    out[0] = loss_ws[0] + 0.1f /*ALPHA*/ * mx;
}

// ---------------- launcher ------------------------------------------
extern "C" void kernel_launch(void* const* d_in, const int* in_sizes, int n_in,
                              void* d_out, int out_size, void* d_ws, size_t ws_size,
                              hipStream_t stream) {
    const float* data = (const float*)d_in[0];      // (64, 512, 64) fp32
    const int*   lens = (const int*)d_in[1];        // (64,) int32
    float* out = (float*)d_out;                     // scalar fp32
    float* ws  = (float*)d_ws;

    // workspace layout (floats): norms | Dm | raw | mmds | loss  (~63 MB)
    float* norms = ws;                                        // 64*512
    float* Dm    = norms + 64 * T_LEN;                        // 60*512*512
    float* raw   = Dm + (size_t)NPAIR * T_LEN * T_LEN;        // 60 (pad 64)
    float* mmds  = raw + 64;                                  // 16
    float* loss  = mmds + 16;                                 // 1

    k_norms    <<<dim3(64),       dim3(256), 0, stream>>>(data, norms);
    k_cost_wmma<<<dim3(128, 60),  dim3(256), 0, stream>>>(data, norms, Dm);
    k_softdtw  <<<dim3(NPAIR),    dim3(512), 0, stream>>>(Dm, lens, raw);
    k_triplet  <<<dim3(1),        dim3(32),  0, stream>>>(raw, lens, loss);
    k_mmd      <<<dim3(MMD_PAIRS),dim3(256), 0, stream>>>(data, mmds);
    k_final    <<<dim3(1),        dim3(32),  0, stream>>>(loss, mmds, out);
}